// FilteredSemiCRFLoss_48765058679283
// MI455X (gfx1250) — compile-verified
//
#include <hip/hip_runtime.h>
#include <math.h>

#define B_  16
#define L_  180
#define W_  16
#define C_  10
#define N_  2760
#define NEGF (-1e30f)

typedef float v2f __attribute__((ext_vector_type(2)));
typedef float v8f __attribute__((ext_vector_type(8)));

__device__ __forceinline__ float laexp(float a, float b) {
    float m = fmaxf(a, b);
    float d = fminf(a, b) - m;           // <= 0; both NEG -> 0 -> m + ln2 (matches jnp.logaddexp)
    return m + log1pf(expf(d));
}

__global__ __launch_bounds__(32) void semicrf_kernel(
    const int*   __restrict__ labels,   // (B,N) segment labels (v)
    const int*   __restrict__ filt,     // (B,N) filter labels (l)
    const float* __restrict__ scores,   // (B,N) scores (g)
    const float* __restrict__ T,        // (C,C) transitions
    float*       __restrict__ perBatch) // (B) losses
{
    __shared__ float sT[C_ * C_];
    __shared__ float sE[12 * 16];       // exp(T), padded to K=12 rows, 16 cols
    __shared__ float sG[N_];
    __shared__ int   sMeta[N_];         // l | gold<<8 | potval<<9 | cand<<10
    __shared__ int   sCov[L_];
    __shared__ int   sPref[L_ + 1];
    __shared__ float sPend[L_ * C_];    // pending[end][label] = lse of alphas
    __shared__ float sLse[C_];
    __shared__ float sU[12];            // exp(LSE - mL), rows 10,11 stay 0
    __shared__ float sRed[32];
    __shared__ int   sCnt[2];           // n_gold, n_cand
    __shared__ float sGoldScore;

    const int b    = blockIdx.x;
    const int lane = threadIdx.x;

    if (lane == 0) { sCnt[0] = 0; sCnt[1] = 0; }
    for (int i = lane; i < C_ * C_; i += 32) sT[i] = T[i];
    for (int p = lane; p < L_; p += 32) sCov[p] = 0;
    if (lane < 12) sU[lane] = 0.0f;
    __syncthreads();
    for (int i = lane; i < 12 * 16; i += 32) {
        int r = i >> 4, c = i & 15;
        sE[i] = (r < C_ && c < C_) ? expf(sT[r * C_ + c]) : 0.0f;
    }

    // ---- pass 1: stage data, mark gold coverage, count gold ----
    int myGold = 0;
    for (int w = 0; w < W_; ++w) {
        int off = w * L_ - (w * (w - 1)) / 2;
        for (int s = lane; s < L_ - w; s += 32) {
            int j = off + s;
            int v = labels[b * N_ + j];
            int l = filt[b * N_ + j];
            sG[j] = scores[b * N_ + j];
            int gold = (v > 0) ? 1 : 0;
            int potv = (v == 0 && l > 0) ? 1 : 0;
            sMeta[j] = l | (gold << 8) | (potv << 9);
            if (gold) {
                ++myGold;
                for (int p = s; p <= s + w; ++p) sCov[p] = 1;  // benign races
            }
        }
    }
    atomicAdd(&sCnt[0], myGold);
    __syncthreads();
    if (lane == 0) {
        int acc = 0; sPref[0] = 0;
        for (int p = 0; p < L_; ++p) { acc += sCov[p]; sPref[p + 1] = acc; }
    }
    __syncthreads();

    // ---- pass 2: cand flags, count ----
    int myCand = 0;
    for (int w = 0; w < W_; ++w) {
        int off = w * L_ - (w * (w - 1)) / 2;
        for (int s = lane; s < L_ - w; s += 32) {
            int j = off + s;
            int m = sMeta[j];
            int ov   = (sPref[s + w + 1] > sPref[s]) ? 1 : 0;
            int cand = ((m >> 8) & 1) | (((m >> 9) & 1) & ov);
            sMeta[j] = m | (cand << 10);
            myCand += cand;
        }
    }
    atomicAdd(&sCnt[1], myCand);
    for (int i = lane; i < L_ * C_; i += 32) sPend[i] = NEGF;
    if (lane < C_) sLse[lane] = NEGF;
    __syncthreads();

    // ---- gold path score (sorted (s,e) order), scalar chain ----
    if (lane == 0) {
        int prev = 0; float acc = 0.0f;
        for (int s = 0; s < L_; ++s) {
            int wmax = (L_ - 1 - s < W_ - 1) ? (L_ - 1 - s) : (W_ - 1);
            for (int w = 0; w <= wmax; ++w) {
                int j = w * L_ - (w * (w - 1)) / 2 + s;
                int m = sMeta[j];
                if ((m >> 8) & 1) {
                    int l = m & 0xFF;
                    acc += sG[j] + sT[prev * C_ + l];
                    prev = l;
                }
            }
        }
        sGoldScore = acc + sT[prev * C_ + 0];
    }
    __syncthreads();

    // ---- streaming DP over start positions; WMMA matvec per step ----
    const int  n      = lane & 15;          // width handled by this lane
    const bool hi     = lane >= 16;
    const int  k0base = hi ? 2 : 0;         // A/B frag K mapping (16x4 f32 layout)
    const int  jbase  = n * L_ - (n * (n - 1)) / 2;

    for (int s = 0; s < L_; ++s) {
        if (s > 0 && lane < C_)
            sLse[lane] = laexp(sLse[lane], sPend[(s - 1) * C_ + lane]);
        __syncthreads();
        float mL = sLse[0];
        #pragma unroll
        for (int c = 1; c < C_; ++c) mL = fmaxf(mL, sLse[c]);
        if (lane < C_) sU[lane] = expf(sLse[lane] - mL);
        __syncthreads();

        bool exists = (s + n) < L_;
        int  j  = jbase + s;
        int  jc = (j < N_) ? j : 0;
        int  m  = exists ? sMeta[jc] : 0;
        int  ln = m & 0xFF;
        bool cand = ((m >> 10) & 1) != 0;

        // P[w] = sum_c u[c] * E[c][l_w]  via 3 chained V_WMMA_F32_16X16X4_F32 (K=12)
        v8f acc = {};
        #pragma unroll
        for (int ck = 0; ck < 3; ++ck) {
            int k0 = ck * 4 + k0base;
            v2f a;  a[0]  = sU[k0];            a[1]  = sU[k0 + 1];
            v2f bb; bb[0] = sE[k0 * 16 + ln];  bb[1] = sE[(k0 + 1) * 16 + ln];
            acc = __builtin_amdgcn_wmma_f32_16x16x4_f32(false, a, false, bb,
                                                        (short)0, acc, false, false);
        }
        float P = acc[0];                   // all D rows identical; col = lane&15

        if (!hi && cand) {
            float lsePred = mL + logf(P);
            float aJ = sG[j] + laexp(sT[ln], lsePred);      // sT[ln] = T[0][l]
            int pe = (s + n) * C_ + ln;
            sPend[pe] = laexp(sPend[pe], aJ);
        }
        __syncthreads();
    }

    // ---- log_z = logsumexp over (end,label) of pending + T[l][0] ----
    float mymax = NEGF;
    for (int i = lane; i < L_ * C_; i += 32) {
        int l = i % C_;
        mymax = fmaxf(mymax, sPend[i] + sT[l * C_ + 0]);
    }
    sRed[lane] = mymax;
    __syncthreads();
    float M = sRed[0];
    for (int i = 1; i < 32; ++i) M = fmaxf(M, sRed[i]);
    float mysum = 0.0f;
    for (int i = lane; i < L_ * C_; i += 32) {
        int l = i % C_;
        mysum += expf(sPend[i] + sT[l * C_ + 0] - M);
    }
    __syncthreads();
    sRed[lane] = mysum;
    __syncthreads();
    if (lane == 0) {
        float S = 0.0f;
        for (int i = 0; i < 32; ++i) S += sRed[i];   // fixed order: deterministic
        float logZ = M + logf(S);
        int nGold = sCnt[0], nCand = sCnt[1];
        perBatch[b] = (nGold < 1 || nCand == nGold) ? 0.0f : (logZ - sGoldScore);
    }
}

__global__ void sum16_kernel(const float* __restrict__ perBatch,
                             float* __restrict__ out) {
    if (blockIdx.x == 0 && threadIdx.x == 0) {
        float s = 0.0f;
        for (int i = 0; i < B_; ++i) s += perBatch[i];
        out[0] = s;
    }
}

extern "C" void kernel_launch(void* const* d_in, const int* in_sizes, int n_in,
                              void* d_out, int out_size, void* d_ws, size_t ws_size,
                              hipStream_t stream) {
    // inputs: 0=all_segment_idx (unused; spans are analytic), 1=labels, 2=filter,
    //         3=scores, 4=transition
    const int*   labels = (const int*)  d_in[1];
    const int*   filt   = (const int*)  d_in[2];
    const float* scores = (const float*)d_in[3];
    const float* T      = (const float*)d_in[4];
    float* perBatch = (float*)d_ws;

    semicrf_kernel<<<dim3(B_), dim3(32), 0, stream>>>(labels, filt, scores, T, perBatch);
    sum16_kernel<<<dim3(1), dim3(32), 0, stream>>>(perBatch, (float*)d_out);
}